// NeighborCooccurrenceEncoder_80333068304563
// MI455X (gfx1250) — compile-verified
//
#include <hip/hip_runtime.h>
#include <hip/hip_bf16.h>

// ---------------------------------------------------------------------------
// NeighborCooccurrenceEncoder for MI455X (gfx1250, wave32)
//
// Shapes: B=4096, N=128, F=64.
// Insight: MLP input is an integer co-occurrence count t in [0,128]
// (padding == t=0 case), so  out_row(t) = w2 @ relu(t*w1 + b1) + b2
// is a 129-entry x 64-float lookup table. Build it ONCE with f32 WMMA
// (exact, matrix-pipe rate), then the main kernel is purely bandwidth
// bound: 256 MB of coalesced nontemporal stores (~11 us @ 23.3 TB/s).
// The 33 KB table is streamed global->LDS with CDNA5 async-LDS copies,
// overlapped with the VALU co-occurrence count phase.
// ---------------------------------------------------------------------------

typedef __attribute__((ext_vector_type(2))) float v2f;
typedef __attribute__((ext_vector_type(4))) float v4f;
typedef __attribute__((ext_vector_type(8))) float v8f;
typedef __attribute__((ext_vector_type(4))) int   v4i;
typedef int b128_t __attribute__((vector_size(16)));   // builtin's pointee type

#define B_  4096
#define N_  128
#define F_  64
#define NT  129          // table entries: counts 0..128
#define NTP 144          // padded to 9 WMMA row-tiles

#if defined(__has_builtin)
#if __has_builtin(__builtin_amdgcn_global_load_async_to_lds_b128) && \
    __has_builtin(__builtin_amdgcn_s_wait_asynccnt)
#define HAVE_ASYNC_LDS 1
#endif
#endif
#ifndef HAVE_ASYNC_LDS
#define HAVE_ASYNC_LDS 0
#endif

#define AS_GLOBAL __attribute__((address_space(1)))
#define AS_LDS    __attribute__((address_space(3)))

// ---------------------------------------------------------------------------
// Kernel 1: phi[t][g] = sum_f w2[g][f] * relu(t*w1[f]+b1[f]) + b2[g]
// One wave (32 threads) per 16x16 output tile, grid = (9 M-tiles, 4 N-tiles).
// K=64 reduced with 16 steps of V_WMMA_F32_16X16X4_F32 (exact f32).
// ---------------------------------------------------------------------------
__global__ __launch_bounds__(32) void build_phi_kernel(
    const float* __restrict__ w1, const float* __restrict__ b1,
    const float* __restrict__ w2, const float* __restrict__ b2,
    float* __restrict__ phi)
{
    const int mt   = blockIdx.x;           // 0..8  (rows of table, 16 t's each)
    const int nt   = blockIdx.y;           // 0..3  (16 g's each)
    const int lane = threadIdx.x;          // 0..31, full wave (EXEC all ones)
    const int lo16 = lane & 15;
    const int hi   = lane >> 4;            // 0 or 1

    const float t = (float)(mt * 16 + lo16);   // A-matrix row for this lane
    const int   g = nt * 16 + lo16;            // B/C/D column for this lane

    v8f c = {};
    #pragma unroll
    for (int k = 0; k < 16; ++k) {
        // This k-step covers K (=f) indices k*4 .. k*4+3.
        // A layout (16x4 f32): lane holds M=lane%16, K = (lane/16)*2 + {0,1}
        // B layout (4x16 f32): lane holds N=lane%16, K = (lane/16)*2 + {0,1}
        const int f0 = k * 4 + hi * 2;
        const int f1 = f0 + 1;
        v2f a, bb;
        a[0]  = fmaxf(t * w1[f0] + b1[f0], 0.0f);   // relu(t*w1+b1)
        a[1]  = fmaxf(t * w1[f1] + b1[f1], 0.0f);
        bb[0] = w2[g * F_ + f0];                    // B = w2^T (einsum 'f,gf->g')
        bb[1] = w2[g * F_ + f1];
        c = __builtin_amdgcn_wmma_f32_16x16x4_f32(
                /*neg_a=*/false, a, /*neg_b=*/false, bb,
                /*c_mod=*/(short)0, c, /*reuse_a=*/false, /*reuse_b=*/false);
    }

    // C/D layout: VGPR r holds M = r + (lane/16)*8, N = lane%16
    const float bias = b2[g];
    #pragma unroll
    for (int r = 0; r < 8; ++r) {
        const int trow = mt * 16 + r + hi * 8;
        phi[trow * F_ + g] = c[r] + bias;
    }
}

// ---------------------------------------------------------------------------
// Kernel 2: per-batch counts + table gather + streamed output.
// 1 block per batch, 128 threads (4 waves). ~36.5 KB LDS per block
// (~8 blocks per 320 KB WGP for latency hiding).
// ---------------------------------------------------------------------------
__global__ __launch_bounds__(128) void cooc_main_kernel(
    const int* __restrict__ src_ids, const int* __restrict__ dst_ids,
    const float* __restrict__ phi_g, float* __restrict__ out)
{
    __shared__ v4i s_src4[N_ / 4];
    __shared__ v4i s_dst4[N_ / 4];
    __shared__ int s_i0[2][N_];            // [side][row] channel-0 table index
    __shared__ int s_i1[2][N_];            // [side][row] channel-1 table index
    __shared__ v4f s_phi4[NT * (F_ / 4)];  // 129 x 64 floats, 16B aligned

    int*   s_src = (int*)s_src4;
    int*   s_dst = (int*)s_dst4;

    const int b   = blockIdx.x;
    const int tid = threadIdx.x;

    // ---- Kick off async staging of the phi table (33 KB, L2-resident). ---
    // GLOBAL_LOAD_ASYNC_TO_LDS_B128: no VGPR data path, tracked by ASYNCcnt;
    // completes while we run the count phase below.
    const v4f* phi_g4 = (const v4f*)phi_g;
#if HAVE_ASYNC_LDS
    for (int i = tid; i < NT * (F_ / 4); i += 128) {
        // two-step casts: (const-cast + reinterpret) in generic AS, then
        // explicit addrspacecast to the builtin's typed AS1/AS3 pointers.
        b128_t* gp_gen = (b128_t*)(phi_g4 + i);
        b128_t* lp_gen = (b128_t*)(&s_phi4[i]);
        __builtin_amdgcn_global_load_async_to_lds_b128(
            (AS_GLOBAL b128_t*)gp_gen,
            (AS_LDS    b128_t*)lp_gen,
            /*offset=*/0, /*cpol=*/0);
    }
#else
    for (int i = tid; i < NT * (F_ / 4); i += 128) s_phi4[i] = phi_g4[i];
#endif

    // ids: synchronous load -> LDS (needed immediately by the count phase)
    s_src[tid] = src_ids[(size_t)b * N_ + tid];
    s_dst[tid] = dst_ids[(size_t)b * N_ + tid];
    __syncthreads();   // ids visible (phi may still be in flight)

    // --- Pairwise counts via LDS-broadcast compares (b128 LDS reads) ------
    const int ms = s_src[tid], md = s_dst[tid];
    int css = 0, csd = 0, cdd = 0, cds = 0;
    #pragma unroll 4
    for (int j4 = 0; j4 < N_ / 4; ++j4) {
        const v4i sj = s_src4[j4];
        const v4i dj = s_dst4[j4];
        #pragma unroll
        for (int e = 0; e < 4; ++e) {
            css += (ms == sj[e]); csd += (ms == dj[e]);
            cdd += (md == dj[e]); cds += (md == sj[e]);
        }
    }
    // Padding (id == -1) zeroes both freq channels -> table index 0,
    // which correctly yields w2@relu(b1)+b2 per channel.
    const bool ps = (ms == -1), pd = (md == -1);
    s_i0[0][tid] = ps ? 0 : css;   // src side: [src_in_src, src_in_dst]
    s_i1[0][tid] = ps ? 0 : csd;
    s_i0[1][tid] = pd ? 0 : cdd;   // dst side: [dst_in_dst, dst_in_src]
    s_i1[1][tid] = pd ? 0 : cds;

#if HAVE_ASYNC_LDS
    __builtin_amdgcn_s_wait_asynccnt(0);   // this wave's phi chunks landed
#endif
    __syncthreads();   // all waves' phi chunks + count indices visible

    // --- Output: out[side][b][r][g] = phi[i0[r]][g] + phi[i1[r]][g] -------
    // 8 rows x 16 lanes x float4 per iteration: 2 KB contiguous stores.
    const int r_off = tid >> 4;          // 0..7
    const int gq    = tid & 15;          // float4 column 0..15
    #pragma unroll
    for (int side = 0; side < 2; ++side) {
        float* outp = out + (size_t)side * B_ * N_ * F_ + (size_t)b * N_ * F_;
        for (int base = 0; base < N_; base += 8) {
            const int r  = base + r_off;
            const int i0 = s_i0[side][r];
            const int i1 = s_i1[side][r];
            const v4f p0 = s_phi4[i0 * (F_ / 4) + gq];
            const v4f p1 = s_phi4[i1 * (F_ / 4) + gq];
            const v4f v  = p0 + p1;
            // 256 MB stream-out > 192 MB L2: keep it nontemporal.
            __builtin_nontemporal_store(v, (v4f*)(outp + r * F_ + gq * 4));
        }
    }
}

// ---------------------------------------------------------------------------
extern "C" void kernel_launch(void* const* d_in, const int* in_sizes, int n_in,
                              void* d_out, int out_size, void* d_ws, size_t ws_size,
                              hipStream_t stream)
{
    const int*   src = (const int*)d_in[0];
    const int*   dst = (const int*)d_in[1];
    const float* w1  = (const float*)d_in[2];
    const float* b1  = (const float*)d_in[3];
    const float* w2  = (const float*)d_in[4];
    const float* b2  = (const float*)d_in[5];
    float*       phi = (float*)d_ws;           // NTP*F_ floats = 36 KB scratch
    float*       out = (float*)d_out;

    build_phi_kernel<<<dim3(9, 4), 32, 0, stream>>>(w1, b1, w2, b2, phi);
    cooc_main_kernel<<<B_, 128, 0, stream>>>(src, dst, phi, out);
}